// EdgeEmbeddingBlock_4544075399634
// MI455X (gfx1250) — compile-verified
//
#include <hip/hip_runtime.h>
#include <hip/hip_bf16.h>

typedef __attribute__((ext_vector_type(2))) float v2f;
typedef __attribute__((ext_vector_type(4))) float v4f;
typedef __attribute__((ext_vector_type(8))) float v8f;

#define E_EDGES 262144
#define NMAXF   8
#define JSPH    16

__global__ __launch_bounds__(256)
void edge_embed_wmma_kernel(const int*   __restrict__ edge_index,   // [2,E], row 0 = sender
                            const float* __restrict__ radial,       // [E,8]
                            const float* __restrict__ edge_attrs,   // [E,16]
                            const float* __restrict__ node_attrs,   // [N,4]
                            const float* __restrict__ Wm,           // [8,8]
                            const float* __restrict__ bias,         // [8]
                            float*       __restrict__ out)          // [2,E,4,8,16]
{
    const int lane = threadIdx.x & 31;
    const int wave = threadIdx.x >> 5;
    const int tile = blockIdx.x * 8 + wave;   // 16 edges per wave-tile
    const int e0   = tile * 16;
    if (e0 >= E_EDGES) return;                // wave-uniform guard -> EXEC all-1s below

    // ---------------- WMMA: rf[16,8] = radial_tile[16,8] @ W^T + b ----------------
    // V_WMMA_F32_16X16X4_F32, K=8 split into two K=4 steps.
    // A layout (16x4 f32): lanes 0-15 rows M, VGPR0=K0 VGPR1=K1; lanes 16-31: K2/K3.
    const int m  = lane & 15;     // edge row within tile
    const int kh = lane >> 4;     // K half selector
    const int n  = lane & 15;     // output feature column (0..7 valid)

    v2f a0 = *(const v2f*)(radial + (size_t)(e0 + m) * NMAXF + 2 * kh);       // K 0..3
    v2f a1 = *(const v2f*)(radial + (size_t)(e0 + m) * NMAXF + 4 + 2 * kh);   // K 4..7

    // B[k][n] = W[n][k]; pad columns n>=8 with zero (branchless, keeps EXEC full).
    const int   nn  = n & 7;
    const float msk = (n < NMAXF) ? 1.0f : 0.0f;
    v2f b0, b1;
    b0.x = Wm[nn * NMAXF + 2 * kh]     * msk;
    b0.y = Wm[nn * NMAXF + 2 * kh + 1] * msk;
    b1.x = Wm[nn * NMAXF + 4 + 2 * kh] * msk;
    b1.y = Wm[nn * NMAXF + 5 + 2 * kh] * msk;

    const float cv = bias[nn] * msk;   // bias folded into accumulator
    v8f c = { cv, cv, cv, cv, cv, cv, cv, cv };

    c = __builtin_amdgcn_wmma_f32_16x16x4_f32(false, a0, false, b0, (short)0, c, false, false);
    c = __builtin_amdgcn_wmma_f32_16x16x4_f32(false, a1, false, b1, (short)0, c, false, false);
    // D layout: VGPR r, lanes 0-15 -> rf[m=r][n=lane]; lanes 16-31 -> rf[m=r+8][n=lane-16]

    // ---------------- Streaming rank-1 expansion: 1.07 GB of NT stores ----------------
    // flat offset within edge: p = k*128 + i*16 + j. Round r, lane l -> p = r*128 + 4l
    //   => k=r, i=l>>2, j0=4*(l&3); each round = one contiguous 512B burst per copy.
    float* outR = out;
    float* outI = out + (size_t)E_EDGES * 512;
    const int iFeat = lane >> 2;          // 0..7
    const int j0    = (lane & 3) * 4;     // 0,4,8,12

#pragma unroll
    for (int mm = 0; mm < 16; ++mm) {
        const int e   = e0 + mm;
        const int snd = edge_index[e];                                  // sender gather
        v4f sa = *(const v4f*)(node_attrs + (size_t)snd * 4);           // broadcast load
        v4f ea = *(const v4f*)(edge_attrs + (size_t)e * JSPH + j0);
        // broadcast rf[mm][iFeat] from WMMA accumulator across the wave
        const float rfi = __shfl(c[mm & 7], ((mm >> 3) << 4) + iFeat, 32);
        const size_t base = (size_t)e * 512 + (size_t)lane * 4;
#pragma unroll
        for (int k = 0; k < 4; ++k) {
            const float s = rfi * sa[k];
            v4f v = ea * s;
            __builtin_nontemporal_store(v, (v4f*)(outR + base + k * 128));
            __builtin_nontemporal_store(v, (v4f*)(outI + base + k * 128));
        }
    }
}

extern "C" void kernel_launch(void* const* d_in, const int* in_sizes, int n_in,
                              void* d_out, int out_size, void* d_ws, size_t ws_size,
                              hipStream_t stream) {
    const int*   edge_index = (const int*)  d_in[0];  // [2,E]
    const float* radial     = (const float*)d_in[1];  // [E,8]
    const float* edge_attrs = (const float*)d_in[2];  // [E,16]
    const float* node_attrs = (const float*)d_in[3];  // [N,4]
    const float* Wm         = (const float*)d_in[4];  // [8,8]
    const float* bias       = (const float*)d_in[5];  // [8]
    float*       out        = (float*)d_out;          // [2,E,4,8,16]

    const int tiles  = E_EDGES / 16;        // 16384 wave-tiles
    const int blocks = tiles / 8;           // 8 waves (256 thr) per block -> 2048
    hipLaunchKernelGGL(edge_embed_wmma_kernel, dim3(blocks), dim3(256), 0, stream,
                       edge_index, radial, edge_attrs, node_attrs, Wm, bias, out);
}